// MultiConvAttn_82179904241699
// MI455X (gfx1250) — compile-verified
//
#include <hip/hip_runtime.h>

// CDNA5 fp32 WMMA: 16x16x4, A/B = 2 VGPRs/lane, C/D = 8 VGPRs/lane (wave32)
typedef float v2f __attribute__((ext_vector_type(2)));
typedef float v8f __attribute__((ext_vector_type(8)));

#define WMMA_F32_16x16x4(a, b, c) \
    __builtin_amdgcn_wmma_f32_16x16x4_f32(false, (a), false, (b), (short)0, (c), false, false)

#define BB 16
#define SS 1024
#define DD 1024
#define NJ 24          // real columns (H*K)
#define NP 32          // padded columns

// Workspace layout (floats):
//   WqT   [NP*DD]        transposed, zero-padded Wq   (rows 24..31 = 0)
//   bqPad [NP]           zero-padded bias
//   ApadT [BB*NP*DD]     per-batch conv weights, transposed (j, d), rows 24..31 = 0
//   zbuf  [BB*SS*NP]     stage-2 output, padded stride
#define OFF_BQ   (NP * DD)
#define OFF_APAD (OFF_BQ + NP)
#define OFF_Z    (OFF_APAD + BB * NP * DD)

// ---------------------------------------------------------------------------
__global__ void mca_init_wq(const float* __restrict__ Wq, const float* __restrict__ bq,
                            float* __restrict__ WqT, float* __restrict__ bqPad)
{
    const int idx = blockIdx.x * blockDim.x + threadIdx.x;   // NP*DD threads
    if (idx < NP * DD) {
        const int n = idx >> 10, d = idx & (DD - 1);
        WqT[idx] = (n < NJ) ? Wq[d * NJ + n] : 0.0f;
    }
    if (idx < NP) bqPad[idx] = (idx < NJ) ? bq[idx] : 0.0f;
}

__global__ void mca_init_apad(float* __restrict__ ApadT)
{
    const int idx = blockIdx.x * blockDim.x + threadIdx.x;   // BB*8*DD threads
    const int b = idx >> 13;
    const int r = idx & 8191;                                // rows 24..31 region
    ApadT[(size_t)b * (NP * DD) + NJ * DD + r] = 0.0f;
}

// ---------------------------------------------------------------------------
// Stage 1: (B*S,1024)x(1024,24) GEMM; epilogue scatters straight into the
// reshaped per-batch conv-weight view ApadT[b][(h*3+k)*1024 + d].
// ---------------------------------------------------------------------------
__global__ void mca_gemm_qw(const float* __restrict__ Q,     // (B*S, D)
                            const float* __restrict__ WqT,   // (NP, D) transposed+padded
                            const float* __restrict__ bqPad, // (NP)
                            float* __restrict__ ApadT)       // (B, NP, D)
{
    const int lane = threadIdx.x & 31;
    const int wave = blockIdx.x * 4 + (threadIdx.x >> 5);    // 0..1023
    const int m0   = wave << 4;
    const int row  = lane & 15;
    const int half = lane >> 4;

    const float* arow = Q   + (size_t)(m0 + row) * DD + 2 * half;
    const float* b0p  = WqT + (size_t)row * DD + 2 * half;   // B cols row / row+16
    const float* b1p  = b0p + 16 * DD;

    v8f c0 = {}, c1 = {};
#pragma unroll 8
    for (int k0 = 0; k0 < DD; k0 += 4) {
        v2f a  = __builtin_nontemporal_load((const v2f*)(arow + k0));  // single-use stream
        v2f b0 = *(const v2f*)(b0p + k0);                              // reused across waves
        v2f b1 = *(const v2f*)(b1p + k0);
        c0 = WMMA_F32_16x16x4(a, b0, c0);
        c1 = WMMA_F32_16x16x4(a, b1, c1);
    }

    const int n0 = row, n1 = 16 + row;
    const float bias0 = bqPad[n0], bias1 = bqPad[n1];
#pragma unroll
    for (int r = 0; r < 8; ++r) {
        const int m = m0 + r + half * 8;                     // C: VGPR r -> M = r + 8*(lane/16)
        const int b = m >> 10, s = m & (SS - 1);
        float* Ab = ApadT + (size_t)b * (NP * DD);
        // flat t = s*24 + j  ->  (h, d, k) of the (H, D, K) reshape
        const int t0 = s * NJ + n0;
        const int h0 = t0 / 3072, rr0 = t0 - h0 * 3072;
        const int d0 = rr0 / 3,   kk0 = rr0 - d0 * 3;
        Ab[(h0 * 3 + kk0) * DD + d0] = c0[r] + bias0;
        if (n1 < NJ) {
            const int t1 = s * NJ + n1;
            const int h1 = t1 / 3072, rr1 = t1 - h1 * 3072;
            const int d1 = rr1 / 3,   kk1 = rr1 - d1 * 3;
            Ab[(h1 * 3 + kk1) * DD + d1] = c1[r] + bias1;
        }
    }
}

// ---------------------------------------------------------------------------
// Stage 2: per-batch (1024,1024)x(1024,24): z[b][s,j] = sum_d key[b,s,d]*A[b][j,d]
// ---------------------------------------------------------------------------
__global__ void mca_gemm_conv(const float* __restrict__ Key,   // (B, S, D)
                              const float* __restrict__ ApadT, // (B, NP, D)
                              float* __restrict__ zbuf)        // (B, S, NP)
{
    const int lane = threadIdx.x & 31;
    const int wave = blockIdx.x * 4 + (threadIdx.x >> 5);    // 0..1023
    const int b    = wave >> 6;
    const int s0   = (wave & 63) << 4;
    const int row  = lane & 15;
    const int half = lane >> 4;

    const float* arow = Key   + ((size_t)b * SS + s0 + row) * DD + 2 * half;
    const float* b0p  = ApadT + (size_t)b * (NP * DD) + (size_t)row * DD + 2 * half;
    const float* b1p  = b0p + 16 * DD;

    v8f c0 = {}, c1 = {};
#pragma unroll 8
    for (int k0 = 0; k0 < DD; k0 += 4) {
        v2f a  = __builtin_nontemporal_load((const v2f*)(arow + k0));  // single-use stream
        v2f b0 = *(const v2f*)(b0p + k0);
        v2f b1 = *(const v2f*)(b1p + k0);
        c0 = WMMA_F32_16x16x4(a, b0, c0);
        c1 = WMMA_F32_16x16x4(a, b1, c1);
    }

    float* z = zbuf + (size_t)b * (SS * NP);
#pragma unroll
    for (int r = 0; r < 8; ++r) {
        const int s = s0 + r + half * 8;
        z[s * NP + row]      = c0[r];                        // cols 0..15
        z[s * NP + 16 + row] = c1[r];                        // cols 16..31 (24.. are exact 0)
    }
}

// ---------------------------------------------------------------------------
// Stage 3: out[b,h,s] = z[s-1, h*3] + z[s, h*3+1] + z[s+1, h*3+2] + bias[h]
// ---------------------------------------------------------------------------
__global__ void mca_combine(const float* __restrict__ zbuf,
                            const float* __restrict__ conv_bias,
                            float* __restrict__ out)
{
    const int idx = blockIdx.x * blockDim.x + threadIdx.x;   // B*H*S
    if (idx >= BB * 8 * SS) return;
    const int s = idx & (SS - 1);
    const int h = (idx >> 10) & 7;
    const int b = idx >> 13;
    const float* z = zbuf + (size_t)b * (SS * NP);
    float acc = conv_bias[h];
    if (s > 0)      acc += z[(s - 1) * NP + h * 3 + 0];
    acc += z[s * NP + h * 3 + 1];
    if (s < SS - 1) acc += z[(s + 1) * NP + h * 3 + 2];
    out[idx] = acc;
}

// ---------------------------------------------------------------------------
extern "C" void kernel_launch(void* const* d_in, const int* in_sizes, int n_in,
                              void* d_out, int out_size, void* d_ws, size_t ws_size,
                              hipStream_t stream) {
    const float* Q    = (const float*)d_in[0];   // query_sequences (B,S,D)
    const float* Keyp = (const float*)d_in[1];   // key_sequences   (B,S,D)
    const float* Wq   = (const float*)d_in[3];   // (D, 24)
    const float* bq   = (const float*)d_in[4];   // (24)
    const float* cb   = (const float*)d_in[5];   // conv_bias (8)
    (void)in_sizes; (void)n_in; (void)out_size; (void)ws_size;

    float* ws    = (float*)d_ws;
    float* WqT   = ws;
    float* bqPad = ws + OFF_BQ;
    float* ApadT = ws + OFF_APAD;
    float* zbuf  = ws + OFF_Z;

    mca_init_wq  <<<(NP * DD + 255) / 256, 256, 0, stream>>>(Wq, bq, WqT, bqPad);
    mca_init_apad<<<(BB * 8 * DD + 255) / 256, 256, 0, stream>>>(ApadT);
    mca_gemm_qw  <<<256, 128, 0, stream>>>(Q, WqT, bqPad, ApadT);
    mca_gemm_conv<<<256, 128, 0, stream>>>(Keyp, ApadT, zbuf);
    mca_combine  <<<(BB * 8 * SS + 255) / 256, 256, 0, stream>>>(zbuf, cb, (float*)d_out);
}